// BloomWisard_43233140801688
// MI455X (gfx1250) — compile-verified
//
#include <hip/hip_runtime.h>
#include <hip/hip_bf16.h>

// ---------------------------------------------------------------------------
// BloomWISARD on MI455X (gfx1250)
//
// Bit-pack samples (134MB -> 4MB, transposed) and filters (335MB -> 10MB)
// into workspace so the random-access working set is L2-resident.  Main
// kernel: thread = one sample; sample bit-tile + triple-buffered 8KB bloom
// rows staged into LDS with gfx1250 async-to-LDS loads (ASYNCcnt /
// s_wait_asynccnt, depth-2 lookahead); hashes built with scalar-broadcast
// (v_readlane) bit positions and carry-free packed 16|16-bit XORs.
//
// Bit convention (both packed arrays): element e -> word ((e>>7)*4 + (e&3)),
// bit ((e>>2)&31).  This lets the pack kernels use b128 loads + 4 ballots.
//
// WMMA is not applicable: XOR hashing is GF(2)-linear and the bloom tests
// are data-dependent gathers.
// ---------------------------------------------------------------------------

namespace {
constexpr int B_     = 8192;   // samples
constexpr int ENTRY_ = 4096;   // input bits per sample
constexpr int C_     = 10;     // classes
constexpr int T_     = 32;     // tuple size
constexpr int N_     = 128;    // neurons = ENTRY/T
constexpr int F_     = 65536;  // bloom filter size (bits)
constexpr int SW_    = ENTRY_ / 32;  // 128 sample bit-words per sample
constexpr int FW_    = F_ / 32;      // 2048 filter bit-words (8KB row)
constexpr int SB_    = 128;    // samples per block (== blockDim.x)
}

// ---- gfx1250 async global->LDS helpers (CDNA5 ASYNCcnt path) --------------
__device__ __forceinline__ void async_copy16(void* lds_dst, const void* g_src) {
  // VDST = LDS byte address (low 32 bits of the generic shared pointer),
  // VADDR = 64-bit global address.
  asm volatile("global_load_async_to_lds_b128 %0, %1, off"
               :: "v"((unsigned)(unsigned long long)lds_dst), "v"(g_src)
               : "memory");
}
__device__ __forceinline__ void wait_async_le(void) {  // all done
  asm volatile("s_wait_asynccnt 0" ::: "memory");
}
__device__ __forceinline__ void wait_async_le4(void) { // oldest (cnt-4) done
  asm volatile("s_wait_asynccnt 4" ::: "memory");
}

// ---- kernel 1: pack samples into transposed bit matrix --------------------
// packT[word][b], word/bit per the convention above.  b128 loads, 4 ballots.
__global__ void pack_samples_kernel(const int4* __restrict__ samples4,
                                    unsigned* __restrict__ packT) {
  const int b    = blockIdx.x;
  const int tid  = threadIdx.x;          // 256 threads
  const int lane = tid & 31;
  const int wave = tid >> 5;
  const int4* row = samples4 + (size_t)b * (ENTRY_ / 4);
#pragma unroll
  for (int it = 0; it < ENTRY_ / 1024; ++it) {          // 4 iterations
    const int4 v = row[it * 256 + tid];                 // e0 = it*1024+tid*4
    const unsigned m0 = (unsigned)__ballot(v.x != 0);
    const unsigned m1 = (unsigned)__ballot(v.y != 0);
    const unsigned m2 = (unsigned)__ballot(v.z != 0);
    const unsigned m3 = (unsigned)__ballot(v.w != 0);
    if (lane == 0) {
      const int g = it * 8 + wave;                      // e>>7 group
      packT[(size_t)(g * 4 + 0) * B_ + b] = m0;
      packT[(size_t)(g * 4 + 1) * B_ + b] = m1;
      packT[(size_t)(g * 4 + 2) * B_ + b] = m2;
      packT[(size_t)(g * 4 + 3) * B_ + b] = m3;
    }
  }
}

// ---- kernel 2: pack filters into bit rows ---------------------------------
// packF[(c*N+n)][w]; b128 loads, 4 ballots, single lane-0 b128 store.
__global__ void pack_filters_kernel(const float4* __restrict__ filters4,
                                    uint4* __restrict__ packF4) {
  const int rowi = blockIdx.x;           // c*N + n  (1280 rows)
  const int tid  = threadIdx.x;          // 256 threads
  const int lane = tid & 31;
  const int wave = tid >> 5;
  const float4* src = filters4 + (size_t)rowi * (F_ / 4);
  uint4* dst        = packF4 + (size_t)rowi * (FW_ / 4);
  for (int it = 0; it < F_ / 1024; ++it) {              // 64 iterations
    const float4 v = src[it * 256 + tid];
    const unsigned m0 = (unsigned)__ballot(v.x != 0.0f);
    const unsigned m1 = (unsigned)__ballot(v.y != 0.0f);
    const unsigned m2 = (unsigned)__ballot(v.z != 0.0f);
    const unsigned m3 = (unsigned)__ballot(v.w != 0.0f);
    if (lane == 0) {
      uint4 o; o.x = m0; o.y = m1; o.z = m2; o.w = m3;
      dst[it * 8 + wave] = o;                           // words g*4 .. g*4+3
    }
  }
}

// ---- kernel 3: main compute ----------------------------------------------
// grid = (B/SB, C); block = SB threads (4 waves); thread = one sample.
// LDS: sample bit-tile [SW_][SB_] (64KB) + triple-buffered filter row (24KB).
__global__ __launch_bounds__(SB_) void wisard_main_kernel(
    const unsigned* __restrict__ packT,
    const unsigned* __restrict__ packF,
    const int* __restrict__ tuple_mapping,
    const int* __restrict__ hash_matrix,
    float* __restrict__ out) {
  extern __shared__ unsigned lds[];
  unsigned* smp  = lds;                 // SW_*SB_ = 16384 words
  unsigned* fbuf = lds + SW_ * SB_;     // 3 * FW_ = 6144 words

  const int tid  = threadIdx.x;
  const int lane = tid & 31;
  const int b0   = blockIdx.x * SB_;
  const int c    = blockIdx.y;

  // Lane j holds the two packed (16|16-bit) hash words for tuple position j:
  // F=65536 => hash values fit in 16 bits; XOR has no carries.
  const unsigned hm01 = (unsigned)hash_matrix[lane] |
                        ((unsigned)hash_matrix[32 + lane] << 16);
  const unsigned hm23 = (unsigned)hash_matrix[64 + lane] |
                        ((unsigned)hash_matrix[96 + lane] << 16);

  // ---- async-stage the sample bit tile: smp[word][local_b] ----
#pragma unroll 4
  for (int k = 0; k < (SW_ * SB_ / 4) / SB_; ++k) {     // 32 chunks/thread
    const int q    = tid + k * SB_;
    const int word = q >> 5;                            // q / (SB_/4)
    const int lb0  = (q & 31) * 4;
    async_copy16(&smp[q * 4], packT + (size_t)word * B_ + b0 + lb0);
  }
  // ---- async-stage filter rows n=0,1 (depth-2 lookahead) ----
#pragma unroll
  for (int r = 0; r < 2; ++r) {
    const unsigned* rowp = packF + (size_t)(c * N_ + r) * FW_;
    unsigned* buf = fbuf + r * FW_;
#pragma unroll
    for (int k = 0; k < (FW_ / 4) / SB_; ++k) {         // 4 chunks/thread
      const int q = tid + k * SB_;
      async_copy16(&buf[q * 4], rowp + q * 4);
    }
  }

  int acc = 0;
  for (int n = 0; n < N_; ++n) {
    // Async loads complete in order: <=4 outstanding means row n (and the
    // tile) are done; only row n+1 may still be in flight.  On the final
    // iteration nothing younger exists, so wait for 0.
    if (n + 1 < N_) wait_async_le4(); else wait_async_le();
    // After this barrier: row n fully in LDS, and all waves finished
    // computing on buffer (n+2)%3 == (n-1)%3 -> safe to refill it.
    __syncthreads();

    if (n + 2 < N_) {
      const unsigned* rowp = packF + (size_t)(c * N_ + (n + 2)) * FW_;
      unsigned* nb = fbuf + ((n + 2) % 3) * FW_;
#pragma unroll
      for (int k = 0; k < (FW_ / 4) / SB_; ++k) {
        const int q = tid + k * SB_;
        async_copy16(&nb[q * 4], rowp + q * 4);
      }
    }
    if (n + 4 < N_)  // global_prefetch_b8 for the tuple row 4 neurons ahead
      __builtin_prefetch(tuple_mapping + (size_t)c * ENTRY_ + (n + 4) * T_ + lane, 0, 1);

    const unsigned* fb = fbuf + (n % 3) * FW_;

    // lane j of each wave holds bit-position j of this neuron's tuple
    const unsigned pv =
        (unsigned)tuple_mapping[(size_t)c * ENTRY_ + n * T_ + lane];

    unsigned h01 = 0, h23 = 0;
#pragma unroll
    for (int j = 0; j < T_; ++j) {
      const unsigned p  = (unsigned)__builtin_amdgcn_readlane((int)pv, j); // SGPR
      const unsigned sw = ((p >> 7) << 2) | (p & 3u);                      // scalar
      const unsigned w  = smp[sw * SB_ + tid];          // coalesced LDS read
      const unsigned bp = (p >> 2) & 31u;               // scalar
      const unsigned m  = (unsigned)(((int)(w << (31 - bp))) >> 31);
      h01 ^= ((unsigned)__builtin_amdgcn_readlane((int)hm01, j)) & m;
      h23 ^= ((unsigned)__builtin_amdgcn_readlane((int)hm23, j)) & m;
    }

    const unsigned h0 = h01 & 0xffffu, h1 = h01 >> 16;
    const unsigned h2 = h23 & 0xffffu, h3 = h23 >> 16;
    const unsigned t0 = fb[((h0 >> 7) << 2) | (h0 & 3u)] >> ((h0 >> 2) & 31u);
    const unsigned t1 = fb[((h1 >> 7) << 2) | (h1 & 3u)] >> ((h1 >> 2) & 31u);
    const unsigned t2 = fb[((h2 >> 7) << 2) | (h2 & 3u)] >> ((h2 >> 2) & 31u);
    const unsigned t3 = fb[((h3 >> 7) << 2) | (h3 & 3u)] >> ((h3 >> 2) & 31u);
    acc += (int)(t0 & t1 & t2 & t3 & 1u);
  }

  out[(size_t)(b0 + tid) * C_ + c] = (float)acc;
}

// ---------------------------------------------------------------------------
extern "C" void kernel_launch(void* const* d_in, const int* in_sizes, int n_in,
                              void* d_out, int out_size, void* d_ws, size_t ws_size,
                              hipStream_t stream) {
  (void)in_sizes; (void)n_in; (void)out_size; (void)ws_size;
  const int*   tuple_mapping = (const int*)d_in[1];   // (C, ENTRY) int32
  const int*   hash_matrix   = (const int*)d_in[2];   // (H, T)     int32
  const int4*  samples4      = (const int4*)d_in[0];  // (B, ENTRY) int32
  const float4* filters4     = (const float4*)d_in[3];// (C, N, F)  float32
  float*       out           = (float*)d_out;         // (B, C)     float32

  unsigned* packT = (unsigned*)d_ws;                  // 128*8192 words = 4MB
  unsigned* packF = packT + (size_t)SW_ * B_;         // 1280*2048 words = 10MB

  pack_samples_kernel<<<B_, 256, 0, stream>>>(samples4, packT);
  pack_filters_kernel<<<C_ * N_, 256, 0, stream>>>(filters4, (uint4*)packF);

  dim3 grid(B_ / SB_, C_);                            // (64, 10)
  size_t lds_bytes = (size_t)(SW_ * SB_ + 3 * FW_) * sizeof(unsigned); // 88KB
  wisard_main_kernel<<<grid, SB_, lds_bytes, stream>>>(
      packT, packF, tuple_mapping, hash_matrix, out);
}